// CrossAttentionBridge_90623809946215
// MI455X (gfx1250) — compile-verified
//
#include <hip/hip_runtime.h>
#include <math.h>

// Problem constants (match reference)
#define D_MODEL 2048
#define D_H     512
#define B_      4
#define T_      4096

typedef __attribute__((ext_vector_type(2))) float v2f;
typedef __attribute__((ext_vector_type(8))) float v8f;

// ---------------------------------------------------------------------------
// Zero-pad a rows x K matrix into a 16 x K matrix (rows..15 = 0) so the WMMA
// GEMM chain never reads out of bounds and needs no divergence around WMMA.
// ---------------------------------------------------------------------------
__global__ void pad16_kernel(const float* __restrict__ src, float* __restrict__ dst,
                             int rows, int K) {
    int i = blockIdx.x * blockDim.x + threadIdx.x;
    int total = 16 * K;
    if (i < total) {
        int r = i / K;
        dst[i] = (r < rows) ? src[i] : 0.0f;
    }
}

// ---------------------------------------------------------------------------
// Out(16 x N) = A16(16 x K) @ W(K x N), all fp32, via V_WMMA_F32_16X16X4_F32.
// One wave per 16-column tile of N; K is consumed 4 at a time.
//
// Per-lane layouts (wave32), per CDNA5 ISA 7.12.2:
//   A 16x4 : lane L: row = L&15; VGPR0/1 hold K = k0+2*(L>>4) + {0,1}
//   B 4x16 : lane L: col = L&15; VGPR0/1 hold K = k0+2*(L>>4) + {0,1}
//   C/D    : VGPR r: lanes 0-15 -> row r, lanes 16-31 -> row 8+r, col = L&15
// ---------------------------------------------------------------------------
__global__ void gemm16_f32_wmma(const float* __restrict__ A16,
                                const float* __restrict__ W,
                                float* __restrict__ Out,
                                int K, int N) {
    const int lane = threadIdx.x & 31;
    const int wave = threadIdx.x >> 5;
    const int tile = blockIdx.x * (blockDim.x >> 5) + wave;
    const int n0   = tile * 16;

    const int row  = lane & 15;     // A row / B,C,D column within tile
    const int hi   = lane >> 4;     // 0: K-pair {0,1}; 1: K-pair {2,3}
    const int ncol = n0 + row;

    const float* ap = A16 + (size_t)row * K + hi * 2;
    const float* wp = W + (size_t)(hi * 2) * N + ncol;

    v8f acc = {};
    #pragma unroll 8
    for (int k0 = 0; k0 < K; k0 += 4) {
        v2f a;
        a.x = ap[0];
        a.y = ap[1];
        ap += 4;
        v2f b;
        b.x = wp[0];
        b.y = wp[(size_t)N];
        wp += (size_t)4 * N;
        // D = A(16x4) * B(4x16) + C   (fp32 matrix pipe, RNE)
        acc = __builtin_amdgcn_wmma_f32_16x16x4_f32(
            /*neg_a=*/false, a, /*neg_b=*/false, b,
            /*c_mod=*/(short)0, acc, /*reuse_a=*/false, /*reuse_b=*/false);
    }

    // Store the full 16x16 tile: lanes 0-15 rows 0..7, lanes 16-31 rows 8..15.
    float* op = Out + (size_t)(hi * 8) * N + ncol;
    #pragma unroll
    for (int r = 0; r < 8; ++r)
        op[(size_t)r * N] = acc[r];
}

// ---------------------------------------------------------------------------
// out[b,t,:] = last_hidden[b,t,:] + sigmoid(gate) * o[b,:]
// Pure streaming: 128 MiB in + 128 MiB out; float4 (b128) accesses.
// ---------------------------------------------------------------------------
__global__ void axpy_bias_kernel(const float4* __restrict__ x,
                                 const float* __restrict__ o,
                                 const float* __restrict__ gate,
                                 float4* __restrict__ out, int nvec) {
    const float s = 1.0f / (1.0f + __expf(-gate[0]));
    const int stride = gridDim.x * blockDim.x;
    for (int i = blockIdx.x * blockDim.x + threadIdx.x; i < nvec; i += stride) {
        const int b  = i >> 21;                 // (T*D)/4 = 2^21 float4s per batch
        const int dv = i & (D_MODEL / 4 - 1);   // float4 index within a D row
        const float4 xv = x[i];
        const float4 ov = ((const float4*)(o + (size_t)b * D_MODEL))[dv];
        out[i] = make_float4(fmaf(s, ov.x, xv.x),
                             fmaf(s, ov.y, xv.y),
                             fmaf(s, ov.z, xv.z),
                             fmaf(s, ov.w, xv.w));
    }
}

// ---------------------------------------------------------------------------
// Host launcher
// Inputs: 0 last_hidden (B,T,D) | 1 zH (B,1,512) | 2 W_mem (512,2048)
//         3 W_q | 4 W_k | 5 W_v | 6 W_o | 7 gate (scalar)
// W_q / W_k never touched: softmax over KV-len 1 is identically 1.
// ---------------------------------------------------------------------------
extern "C" void kernel_launch(void* const* d_in, const int* in_sizes, int n_in,
                              void* d_out, int out_size, void* d_ws, size_t ws_size,
                              hipStream_t stream) {
    const float* last_hidden = (const float*)d_in[0];
    const float* zH          = (const float*)d_in[1];
    const float* W_mem       = (const float*)d_in[2];
    const float* W_v         = (const float*)d_in[5];
    const float* W_o         = (const float*)d_in[6];
    const float* gate        = (const float*)d_in[7];
    float* out = (float*)d_out;

    // Workspace: Apad 16x512, mem/v/o 16x2048 each (rows 4..15 stay zero).
    float* Apad = (float*)d_ws;
    float* memb = Apad + 16 * D_H;
    float* vbuf = memb + 16 * D_MODEL;
    float* obuf = vbuf + 16 * D_MODEL;

    // 1) Zero-pad zH (4 x 512) -> 16 x 512
    pad16_kernel<<<(16 * D_H + 255) / 256, 256, 0, stream>>>(zH, Apad, B_, D_H);

    // 2) GEMM chain on the fp32 matrix pipe.
    const int tiles    = D_MODEL / 16;        // 128 column tiles
    const int threads  = 256;                 // 8 waves / block
    const int gemmGrid = tiles / (threads / 32);  // 16 blocks

    gemm16_f32_wmma<<<gemmGrid, threads, 0, stream>>>(Apad, W_mem, memb, D_H,     D_MODEL);
    gemm16_f32_wmma<<<gemmGrid, threads, 0, stream>>>(memb, W_v,   vbuf, D_MODEL, D_MODEL);
    gemm16_f32_wmma<<<gemmGrid, threads, 0, stream>>>(vbuf, W_o,   obuf, D_MODEL, D_MODEL);

    // 3) Streaming residual-add epilogue (bandwidth bound: ~256 MiB @ 23.3 TB/s).
    const int nvec = (B_ * T_ * D_MODEL) / 4;  // 8,388,608 float4s
    axpy_bias_kernel<<<8192, 256, 0, stream>>>((const float4*)last_hidden, obuf,
                                               gate, (float4*)out, nvec);
}